// QCNNModule_66185446031727
// MI455X (gfx1250) — compile-verified
//
#include <hip/hip_runtime.h>
#include <hip/hip_bf16.h>
#include <math.h>

// ---------------------------------------------------------------------------
// QCNN hybrid net on MI455X (gfx1250), reformulated as 3 GEMMs + tiny setup:
//   angles = x @ fc_w^T + fc_b                  (bandwidth bound, 64 MB read)
//   S[b,j] = prod_q (bit_{7-q}(j)? sin:cos)(angles[b,q]/2)   (real, rank-1)
//   Final  = S @ U(qparams)^T   (complex U built once from basis columns)
//   out    = |Final|^2 @ w2T + post_b,  w2T folds Z-readout + post_w
// GEMMs run on V_WMMA_F32_16X16X4_F32 (wave32 matrix pipe).
// B-operands (U, w2T) are stored pre-packed in WMMA B-fragment order:
//   P[jp][n][2] = { M[2jp][n], M[2jp+1][n] }  -> one global_load_b64 per lane,
// lanes 0..15 cover one contiguous 128B segment.
// ---------------------------------------------------------------------------

#define NQ    8
#define DIM   256
#define NL    3
#define BATCH 32768
#define INF   512
#define OUTF  512
#define SLD   260   // padded LDS row stride (floats): 260 % 64 == 4 -> b64
                    // A-fragment reads hit 64 distinct banks across the wave

typedef __attribute__((ext_vector_type(2))) float v2f;
typedef __attribute__((ext_vector_type(8))) float v8f;

__device__ __forceinline__ v8f wmma_f32_16x16x4(v2f a, v2f b, v8f c) {
  // 8 args: (neg_a, A, neg_b, B, c_mod, C, reuse_a, reuse_b)
  return __builtin_amdgcn_wmma_f32_16x16x4_f32(false, a, false, b,
                                               (short)0, c, false, false);
}

// ---------------------------------------------------------------------------
// Setup 1: build U^T packed (row j = circuit applied to basis vector e_j).
// Qubit q acts on bit (7-q) of the flat amplitude index (reference reshape
// puts qubit 0 on the most-significant axis).
// Packed store: UrP[(j>>1)*2*DIM + k*2 + (j&1)] = U^T[j][k]
// ---------------------------------------------------------------------------
__global__ __launch_bounds__(128)
void build_unitary(const float* __restrict__ qparams,
                   float* __restrict__ UrP, float* __restrict__ UiP) {
  __shared__ float re[DIM], im[DIM];
  const int j = blockIdx.x;
  const int t = threadIdx.x;                 // 0..127 == one amplitude pair
  for (int k = t; k < DIM; k += 128) { re[k] = (k == j) ? 1.f : 0.f; im[k] = 0.f; }
  __syncthreads();

  for (int l = 0; l < NL; ++l) {
    // RX(theta) on each wire: [[c, -i s], [-i s, c]]
    for (int q = 0; q < NQ; ++q) {
      const int bq = 7 - q;
      float sv, cv;
      __sincosf(0.5f * qparams[l * NQ + q], &sv, &cv);
      const int mask = (1 << bq) - 1;
      const int i0 = ((t >> bq) << (bq + 1)) | (t & mask);
      const int i1 = i0 | (1 << bq);
      const float r0 = re[i0], m0 = im[i0], r1 = re[i1], m1 = im[i1];
      re[i0] = cv * r0 + sv * m1;  im[i0] = cv * m0 - sv * r1;
      re[i1] = cv * r1 + sv * m0;  im[i1] = cv * m1 - sv * r0;
      __syncthreads();
    }
    // CNOT ring: (0,1)..(6,7),(7,0): swap target bit where control bit == 1
    for (int e = 0; e < NQ; ++e) {
      const int bc = 7 - e, bt = 7 - ((e + 1) & 7);
      if (t < 64) {
        const int lo = bc < bt ? bc : bt, hb = bc < bt ? bt : bc;
        int v = ((t >> lo) << (lo + 1)) | (t & ((1 << lo) - 1));
        v = ((v >> hb) << (hb + 1)) | (v & ((1 << hb) - 1));
        const int ia = v | (1 << bc);        // control=1, target=0
        const int ib = ia | (1 << bt);       // control=1, target=1
        float tr = re[ia]; re[ia] = re[ib]; re[ib] = tr;
        float ti = im[ia]; im[ia] = im[ib]; im[ib] = ti;
      }
      __syncthreads();
    }
  }
  {
    const size_t base = (size_t)(j >> 1) * (2 * DIM) + (j & 1);
    for (int k = t; k < DIM; k += 128) {
      UrP[base + 2 * k] = re[k];
      UiP[base + 2 * k] = im[k];
    }
  }
}

// ---------------------------------------------------------------------------
// Setup 2: w2T[i,o] = sum_q post_w[o,q] * (1 - 2*bit_{7-q}(i)), packed:
//          w2P[(i>>1)*2*OUTF + o*2 + (i&1)]
// ---------------------------------------------------------------------------
__global__ __launch_bounds__(256)
void build_w2t(const float* __restrict__ post_w, float* __restrict__ w2P) {
  const int idx = blockIdx.x * 256 + threadIdx.x;   // idx = i*OUTF + o
  const int i = idx / OUTF, o = idx % OUTF;
  float acc = 0.f;
#pragma unroll
  for (int q = 0; q < NQ; ++q)
    acc += (((i >> (7 - q)) & 1) ? -1.f : 1.f) * post_w[o * NQ + q];
  w2P[(size_t)(i >> 1) * (2 * OUTF) + 2 * o + (i & 1)] = acc;
}

// ---------------------------------------------------------------------------
// K1: angles = x @ fc_w^T + fc_b. One wave per sample, coalesced float4
// loads, wave32 xor-tree reduction. Pure bandwidth (64 MB of x).
// ---------------------------------------------------------------------------
__global__ __launch_bounds__(256)
void proj_angles(const float* __restrict__ x, const float* __restrict__ fc_w,
                 const float* __restrict__ fc_b, float* __restrict__ angles) {
  const int lane = threadIdx.x & 31;
  const int s = blockIdx.x * 8 + (threadIdx.x >> 5);
  const float4* xv = reinterpret_cast<const float4*>(x + (size_t)s * INF);
  float acc[NQ];
#pragma unroll
  for (int q = 0; q < NQ; ++q) acc[q] = 0.f;
#pragma unroll
  for (int it = 0; it < 4; ++it) {
    const int k4 = it * 32 + lane;            // 128 float4 per row
    const float4 xe = xv[k4];
#pragma unroll
    for (int q = 0; q < NQ; ++q) {
      const float4 we = reinterpret_cast<const float4*>(fc_w + q * INF)[k4];
      acc[q] += xe.x * we.x + xe.y * we.y + xe.z * we.z + xe.w * we.w;
    }
  }
#pragma unroll
  for (int q = 0; q < NQ; ++q) {
    float v = acc[q];
#pragma unroll
    for (int off = 16; off > 0; off >>= 1) v += __shfl_xor(v, off, 32);
    if (lane == q) angles[s * NQ + q] = v + fc_b[q];
  }
}

// ---------------------------------------------------------------------------
// K2 (fused): per 16-sample tile: build S tile -> WMMA GEMM vs Ur^T/Ui^T ->
// probs -> WMMA GEMM vs w2T -> +post_b -> out. 4 waves / block.
// A-frag layout (16x4 f32): lane L: M = L&15; vgpr0/1 hold K = 2*(L>>4)+{0,1}.
// C/D layout: vgpr v, lane L: row = v + 8*(L>>4), col = L&15.
// B-frags: single global_load_b64 from the packed layouts.
// ---------------------------------------------------------------------------
__global__ __launch_bounds__(128)
void circuit_out(const float* __restrict__ angles,
                 const float* __restrict__ UrP, const float* __restrict__ UiP,
                 const float* __restrict__ w2P, const float* __restrict__ post_b,
                 float* __restrict__ out) {
  __shared__ float sS[16 * SLD];              // S tile, later probs tile
  __shared__ float sCh[16 * NQ], sSh[16 * NQ];
  const int tid  = threadIdx.x;
  const int lane = tid & 31;
  const int wave = tid >> 5;                  // 0..3
  const int hi   = lane >> 4;                 // 0/1
  const int col  = lane & 15;
  const int s0   = blockIdx.x * 16;
  const v8f vzero = {0.f, 0.f, 0.f, 0.f, 0.f, 0.f, 0.f, 0.f};

  // cos/sin(angle/2) table, 16 samples x 8 qubits (128 threads -> 1 each)
  {
    const int m = tid >> 3, q = tid & 7;
    float sv, cv;
    __sincosf(0.5f * angles[(s0 + m) * NQ + q], &sv, &cv);
    sCh[m * NQ + q] = cv;
    sSh[m * NQ + q] = sv;
  }
  __syncthreads();

  // Embedding tile S[m][j]; wave w fills columns [64w, 64w+64)
  for (int r = 0; r < 2; ++r) {
    const int j = wave * 64 + lane + r * 32;
#pragma unroll
    for (int m = 0; m < 16; ++m) {
      float f = 1.f;
#pragma unroll
      for (int q = 0; q < NQ; ++q)
        f *= ((j >> (7 - q)) & 1) ? sSh[m * NQ + q] : sCh[m * NQ + q];
      sS[m * SLD + j] = f;
    }
  }
  __syncthreads();

  // Phase A: Final = S x U^T (complex); wave w owns i in [64w, 64w+64)
  const int iBase = wave * 64;
  v8f fr[4], fi[4];
#pragma unroll
  for (int t = 0; t < 4; ++t) { fr[t] = vzero; fi[t] = vzero; }
  for (int j0 = 0; j0 < DIM; j0 += 4) {
    const int jp = (j0 >> 1) + hi;            // packed row-pair index
    const v2f a = *reinterpret_cast<const v2f*>(&sS[col * SLD + j0 + 2 * hi]);
#pragma unroll
    for (int t = 0; t < 4; ++t) {
      const int i = iBase + t * 16 + col;
      const v2f br = *reinterpret_cast<const v2f*>(&UrP[(size_t)jp * (2 * DIM) + 2 * i]);
      const v2f bi = *reinterpret_cast<const v2f*>(&UiP[(size_t)jp * (2 * DIM) + 2 * i]);
      fr[t] = wmma_f32_16x16x4(a, br, fr[t]);
      fi[t] = wmma_f32_16x16x4(a, bi, fi[t]);
    }
  }
  __syncthreads();                            // all waves done reading S

  // probs = |Final|^2 into the (reused) LDS tile
#pragma unroll
  for (int t = 0; t < 4; ++t) {
#pragma unroll
    for (int v = 0; v < 8; ++v) {
      const int row = v + 8 * hi;
      const int i = iBase + t * 16 + col;
      const float pr = fr[t][v], pi = fi[t][v];
      sS[row * SLD + i] = pr * pr + pi * pi;
    }
  }
  __syncthreads();

  // Phase B: out = probs x w2T + post_b; wave w owns o in [128w, 128w+128)
  const int oBase = wave * 128;
  v8f acc[8];
#pragma unroll
  for (int t = 0; t < 8; ++t) acc[t] = vzero;
  for (int j0 = 0; j0 < DIM; j0 += 4) {
    const int jp = (j0 >> 1) + hi;
    const v2f a = *reinterpret_cast<const v2f*>(&sS[col * SLD + j0 + 2 * hi]);
#pragma unroll
    for (int t = 0; t < 8; ++t) {
      const int o = oBase + t * 16 + col;
      const v2f b = *reinterpret_cast<const v2f*>(&w2P[(size_t)jp * (2 * OUTF) + 2 * o]);
      acc[t] = wmma_f32_16x16x4(a, b, acc[t]);
    }
  }
#pragma unroll
  for (int t = 0; t < 8; ++t) {
    const int o = oBase + t * 16 + col;
    const float bias = post_b[o];
#pragma unroll
    for (int v = 0; v < 8; ++v) {
      const int row = v + 8 * hi;
      out[(size_t)(s0 + row) * OUTF + o] = acc[t][v] + bias;
    }
  }
}

// ---------------------------------------------------------------------------
extern "C" void kernel_launch(void* const* d_in, const int* in_sizes, int n_in,
                              void* d_out, int out_size, void* d_ws, size_t ws_size,
                              hipStream_t stream) {
  (void)in_sizes; (void)n_in; (void)out_size; (void)ws_size;
  const float* x       = (const float*)d_in[0];
  const float* fc_w    = (const float*)d_in[1];
  const float* fc_b    = (const float*)d_in[2];
  const float* qparams = (const float*)d_in[3];
  const float* post_w  = (const float*)d_in[4];
  const float* post_b  = (const float*)d_in[5];
  float* out = (float*)d_out;

  // Workspace layout (2 MB total): angles | UrP | UiP | w2P (all packed)
  char* ws = (char*)d_ws;
  float* angles = (float*)ws;                                        // 1 MB
  float* UrP    = (float*)(ws + (size_t)BATCH * NQ * 4);             // 256 KB
  float* UiP    = (float*)(ws + (size_t)BATCH * NQ * 4 + (size_t)DIM * DIM * 4);
  float* w2P    = (float*)(ws + (size_t)BATCH * NQ * 4 + (size_t)2 * DIM * DIM * 4);

  hipLaunchKernelGGL(build_unitary, dim3(DIM), dim3(128), 0, stream,
                     qparams, UrP, UiP);
  hipLaunchKernelGGL(build_w2t, dim3((DIM * OUTF) / 256), dim3(256), 0, stream,
                     post_w, w2P);
  hipLaunchKernelGGL(proj_angles, dim3(BATCH / 8), dim3(256), 0, stream,
                     x, fc_w, fc_b, angles);
  hipLaunchKernelGGL(circuit_out, dim3(BATCH / 16), dim3(128), 0, stream,
                     angles, UrP, UiP, w2P, post_b, out);
}